// RelGraphSAGE_88648124990310
// MI455X (gfx1250) — compile-verified
//
#include <hip/hip_runtime.h>

typedef float v2f __attribute__((ext_vector_type(2)));
typedef float v8f __attribute__((ext_vector_type(8)));

#define N_NODES 100000
#define N_EDGES 800000
#define N_GRAPHS 2000
#define EMB 64
#define REL 16
#define HID 128
#define NCLASS 10
#define K1 80   // agg1 feature dim (64 src emb + 16 rel)

// ---------------- gather / degree ----------------

__global__ void k_embed(const int* __restrict__ tok, const float* __restrict__ emb,
                        float* __restrict__ x) {
  int idx = blockIdx.x * blockDim.x + threadIdx.x;
  if (idx >= N_NODES * EMB) return;
  int node = idx >> 6;
  int f = idx & 63;
  x[idx] = emb[(size_t)tok[node] * EMB + f];
}

__global__ void k_deg(const int* __restrict__ dst, float* __restrict__ deg) {
  int e = blockIdx.x * blockDim.x + threadIdx.x;
  if (e >= N_EDGES) return;
  atomicAdd(&deg[dst[e]], 1.0f);
}

__global__ void k_rdeg(float* __restrict__ deg) {
  int i = blockIdx.x * blockDim.x + threadIdx.x;
  if (i >= N_NODES) return;
  deg[i] = 1.0f / fmaxf(deg[i], 1.0f);
}

// ---------------- edge scatters (L2-resident atomics) ----------------

__global__ void k_scatter1(const int* __restrict__ src, const int* __restrict__ dst,
                           const int* __restrict__ etype, const float* __restrict__ x,
                           const float* __restrict__ rel, float* __restrict__ agg1) {
  long long idx = (long long)blockIdx.x * blockDim.x + threadIdx.x;
  if (idx >= (long long)N_EDGES * K1) return;
  int e = (int)(idx / K1);
  int f = (int)(idx - (long long)e * K1);
  float v;
  if (f < EMB) v = x[(size_t)src[e] * EMB + f];
  else         v = rel[etype[e] * REL + (f - EMB)];
  atomicAdd(&agg1[(size_t)dst[e] * K1 + f], v);
}

__global__ void k_scatter2(const int* __restrict__ src, const int* __restrict__ dst,
                           const float* __restrict__ h, float* __restrict__ agg2) {
  long long idx = (long long)blockIdx.x * blockDim.x + threadIdx.x;
  if (idx >= (long long)N_EDGES * HID) return;
  int e = (int)(idx >> 7);
  int f = (int)(idx & (HID - 1));
  atomicAdd(&agg2[(size_t)dst[e] * HID + f], h[(size_t)src[e] * HID + f]);
}

// ---------------- WMMA GEMMs ----------------
// Block = 16 nodes x 128 cols. 8 waves; wave w owns 16-col tile w.
// A frag (16x4 f32): lanes 0-15 -> K=k0,k0+1 ; lanes 16-31 -> K=k0+2,k0+3 (row = lane&15).
// B frag (4x16 f32): mirrored. C/D: VGPR r -> row r (lanes 0-15) / row r+8 (lanes 16-31).

__global__ __launch_bounds__(256) void k_gemm1(
    const float* __restrict__ agg1, const float* __restrict__ rdeg,
    const float* __restrict__ x, const float* __restrict__ w1_l,
    const float* __restrict__ w1_r, const float* __restrict__ b1,
    float* __restrict__ h) {
  const int tile = blockIdx.x;
  const int wave = threadIdx.x >> 5;
  const int lane = threadIdx.x & 31;
  const int half = lane >> 4;
  const int lr   = lane & 15;
  const int row0 = tile * 16;
  const int col0 = wave * 16;
  const int rowA = row0 + lr;
  const float rd = rdeg[rowA];

  v8f c = {};
  // aggregated-neighbor part: K = 0..79 over w1_l
  for (int k0 = 0; k0 < K1; k0 += 4) {
    int ka = k0 + 2 * half;
    v2f a, b;
    a.x = agg1[(size_t)rowA * K1 + ka]     * rd;
    a.y = agg1[(size_t)rowA * K1 + ka + 1] * rd;
    b.x = w1_l[(size_t)ka       * HID + col0 + lr];
    b.y = w1_l[(size_t)(ka + 1) * HID + col0 + lr];
    c = __builtin_amdgcn_wmma_f32_16x16x4_f32(false, a, false, b, (short)0, c, false, false);
  }
  // root part: K = 0..63 over w1_r
  for (int k0 = 0; k0 < EMB; k0 += 4) {
    int ka = k0 + 2 * half;
    v2f a, b;
    a.x = x[(size_t)rowA * EMB + ka];
    a.y = x[(size_t)rowA * EMB + ka + 1];
    b.x = w1_r[(size_t)ka       * HID + col0 + lr];
    b.y = w1_r[(size_t)(ka + 1) * HID + col0 + lr];
    c = __builtin_amdgcn_wmma_f32_16x16x4_f32(false, a, false, b, (short)0, c, false, false);
  }
  const float bias = b1[col0 + lr];
#pragma unroll
  for (int r = 0; r < 8; ++r) {
    int row = row0 + r + 8 * half;
    h[(size_t)row * HID + col0 + lr] = fmaxf(c[r] + bias, 0.0f);
  }
}

__global__ __launch_bounds__(256) void k_gemm2(
    const float* __restrict__ agg2, const float* __restrict__ rdeg,
    const float* __restrict__ h, const float* __restrict__ w2_l,
    const float* __restrict__ w2_r, const float* __restrict__ b2,
    const int* __restrict__ batch, unsigned int* __restrict__ g) {
  const int tile = blockIdx.x;
  const int wave = threadIdx.x >> 5;
  const int lane = threadIdx.x & 31;
  const int half = lane >> 4;
  const int lr   = lane & 15;
  const int row0 = tile * 16;
  const int col0 = wave * 16;
  const int rowA = row0 + lr;
  const float rd = rdeg[rowA];

  v8f c = {};
  for (int k0 = 0; k0 < HID; k0 += 4) {
    int ka = k0 + 2 * half;
    v2f a, b;
    a.x = agg2[(size_t)rowA * HID + ka]     * rd;
    a.y = agg2[(size_t)rowA * HID + ka + 1] * rd;
    b.x = w2_l[(size_t)ka       * HID + col0 + lr];
    b.y = w2_l[(size_t)(ka + 1) * HID + col0 + lr];
    c = __builtin_amdgcn_wmma_f32_16x16x4_f32(false, a, false, b, (short)0, c, false, false);
  }
  for (int k0 = 0; k0 < HID; k0 += 4) {
    int ka = k0 + 2 * half;
    v2f a, b;
    a.x = h[(size_t)rowA * HID + ka];
    a.y = h[(size_t)rowA * HID + ka + 1];
    b.x = w2_r[(size_t)ka       * HID + col0 + lr];
    b.y = w2_r[(size_t)(ka + 1) * HID + col0 + lr];
    c = __builtin_amdgcn_wmma_f32_16x16x4_f32(false, a, false, b, (short)0, c, false, false);
  }
  const float bias = b2[col0 + lr];
#pragma unroll
  for (int r = 0; r < 8; ++r) {
    int row = row0 + r + 8 * half;
    // fuse segment_max + trailing relu: clamp to >=0, uint-ordered atomic max
    float v = fmaxf(c[r] + bias, 0.0f);
    atomicMax(&g[(size_t)batch[row] * HID + col0 + lr], __float_as_uint(v));
  }
}

// ---------------- output head ----------------

__global__ void k_out(const unsigned int* __restrict__ g, const float* __restrict__ lin_w,
                      const float* __restrict__ lin_b, float* __restrict__ out) {
  int idx = blockIdx.x * blockDim.x + threadIdx.x;
  if (idx >= N_GRAPHS * NCLASS) return;
  int gi = idx / NCLASS;
  int cl = idx - gi * NCLASS;
  float acc = lin_b[cl];
#pragma unroll 8
  for (int k = 0; k < HID; ++k)
    acc += __uint_as_float(g[(size_t)gi * HID + k]) * lin_w[(size_t)k * NCLASS + cl];
  out[idx] = acc;
}

extern "C" void kernel_launch(void* const* d_in, const int* in_sizes, int n_in,
                              void* d_out, int out_size, void* d_ws, size_t ws_size,
                              hipStream_t stream) {
  const int*   x_tokens  = (const int*)d_in[0];
  const int*   edge_idx  = (const int*)d_in[1];
  const int*   edge_type = (const int*)d_in[2];
  const int*   batch     = (const int*)d_in[3];
  // d_in[4] = num_graphs scalar (compile-time N_GRAPHS)
  const float* emb_table = (const float*)d_in[5];
  const float* rel_table = (const float*)d_in[6];
  const float* w1_l      = (const float*)d_in[7];
  const float* b1        = (const float*)d_in[8];
  const float* w1_r      = (const float*)d_in[9];
  const float* w2_l      = (const float*)d_in[10];
  const float* b2        = (const float*)d_in[11];
  const float* w2_r      = (const float*)d_in[12];
  const float* lin_w     = (const float*)d_in[13];
  const float* lin_b     = (const float*)d_in[14];

  const int* src = edge_idx;
  const int* dst = edge_idx + N_EDGES;

  float* ws = (float*)d_ws;
  size_t o = 0;
  float* x    = ws + o; o += (size_t)N_NODES * EMB;
  float* deg  = ws + o; o += (size_t)N_NODES;
  float* agg1 = ws + o; o += (size_t)N_NODES * K1;
  float* h    = ws + o; o += (size_t)N_NODES * HID;
  float* agg2 = ws + o; o += (size_t)N_NODES * HID;
  unsigned int* g = (unsigned int*)(ws + o); o += (size_t)N_GRAPHS * HID;

  hipMemsetAsync(deg,  0, (size_t)N_NODES * sizeof(float), stream);
  hipMemsetAsync(agg1, 0, (size_t)N_NODES * K1 * sizeof(float), stream);
  hipMemsetAsync(agg2, 0, (size_t)N_NODES * HID * sizeof(float), stream);
  hipMemsetAsync(g,    0, (size_t)N_GRAPHS * HID * sizeof(float), stream);

  {
    int total = N_NODES * EMB;
    k_embed<<<(total + 255) / 256, 256, 0, stream>>>(x_tokens, emb_table, x);
  }
  k_deg<<<(N_EDGES + 255) / 256, 256, 0, stream>>>(dst, deg);
  k_rdeg<<<(N_NODES + 255) / 256, 256, 0, stream>>>(deg);
  {
    long long total = (long long)N_EDGES * K1;
    k_scatter1<<<(int)((total + 255) / 256), 256, 0, stream>>>(src, dst, edge_type, x,
                                                               rel_table, agg1);
  }
  k_gemm1<<<N_NODES / 16, 256, 0, stream>>>(agg1, deg, x, w1_l, w1_r, b1, h);
  {
    long long total = (long long)N_EDGES * HID;
    k_scatter2<<<(int)((total + 255) / 256), 256, 0, stream>>>(src, dst, h, agg2);
  }
  k_gemm2<<<N_NODES / 16, 256, 0, stream>>>(agg2, deg, h, w2_l, w2_r, b2, batch, g);
  {
    int total = N_GRAPHS * NCLASS;
    k_out<<<(total + 255) / 256, 256, 0, stream>>>(g, lin_w, lin_b, (float*)d_out);
  }
}